// PolicyNetwork_73684458930386
// MI455X (gfx1250) — compile-verified
//
#include <hip/hip_runtime.h>
#include <hip/hip_bf16.h>

#define N_NODES 50000
#define N_EDGES 800000
#define NODE_F  128
#define EDGE_F  16
#define HID     64
#define N_ACT   6
#define M_TILES (N_NODES / 16)   // 3125 exact

typedef __attribute__((ext_vector_type(2))) float v2f;
typedef __attribute__((ext_vector_type(8))) float v8f;

// ---------------- init: deg=1 (self loop), aggregators/sums = 0 ----------------
__global__ void k_init(float* __restrict__ deg, float* __restrict__ hn_agg,
                       float* __restrict__ he_agg, float* __restrict__ sums) {
    int t = blockIdx.x * blockDim.x + threadIdx.x;
    if (t < N_NODES * HID) { hn_agg[t] = 0.f; he_agg[t] = 0.f; }
    if (t < N_NODES) deg[t] = 1.0f;
    if (t < 2 * HID) sums[t] = 0.f;
}

// ---------------- degree over real edges ----------------
__global__ void k_degree(const int* __restrict__ dstv, float* __restrict__ deg) {
    int e = blockIdx.x * blockDim.x + threadIdx.x;
    if (e < N_EDGES) atomicAdd(&deg[dstv[e]], 1.0f);
}

__global__ void k_rsqrt(float* __restrict__ deg) {
    int i = blockIdx.x * blockDim.x + threadIdx.x;
    if (i < N_NODES) deg[i] = rsqrtf(deg[i]);   // deg >= 1 always (self loop)
}

// ---------------- hn_pre = node_x @ Wn via V_WMMA_F32_16X16X4_F32 ----------------
// 8 waves/block; wave -> one 16x16 C tile. mtile = 2 per block, ntile = HID/16 = 4.
__global__ void k_gemm_node(const float* __restrict__ X, const float* __restrict__ W,
                            float* __restrict__ H) {
    const int lane  = threadIdx.x;           // 0..31
    const int wave  = threadIdx.y;           // 0..7
    const int mtile = blockIdx.x * 2 + (wave >> 2);
    if (mtile >= M_TILES) return;            // uniform per wave: EXEC stays all-ones
    const int ntile = wave & 3;
    const int m     = lane & 15;
    const int khalf = lane >> 4;             // lanes 16-31 hold K+2..K+3 per ISA layout
    const int n     = ntile * 16 + m;

    const float* xrow = X + (mtile * 16 + m) * NODE_F;
    v8f acc = {};
#pragma unroll
    for (int k = 0; k < NODE_F; k += 4) {
        // A (16x4): lane m, VGPR0/1 = K = k + khalf*2 + {0,1}   (8B-aligned)
        v2f a = *(const v2f*)(xrow + k + khalf * 2);
        // B (4x16): lane n, same K split
        v2f b;
        b.x = W[(k + khalf * 2 + 0) * HID + n];
        b.y = W[(k + khalf * 2 + 1) * HID + n];
        acc = __builtin_amdgcn_wmma_f32_16x16x4_f32(
            false, a, false, b, (short)0, acc, false, false);
    }
    // C/D layout: VGPR r -> row khalf*8 + r, col = lane&15 (within tile)
    float* hb = H + (mtile * 16 + khalf * 8) * HID + n;
#pragma unroll
    for (int r = 0; r < 8; ++r) hb[r * HID] = acc[r];
}

// ---------------- edge MLP: he_pre rows < N_NODES stored; rows >= N_NODES ----------------
// directly relu(h+be) -> pe column-sum (deg=1 -> norm=1 self loop only).
__global__ void k_edge_mlp(const float* __restrict__ EX, const float* __restrict__ We,
                           const float* __restrict__ be, float* __restrict__ he_pre,
                           float* __restrict__ pe_sum) {
    __shared__ float sW[EDGE_F * HID];   // 4 KB
    __shared__ float sred[256];
    const int t = threadIdx.x;           // 256 threads
    for (int i = t; i < EDGE_F * HID; i += 256) sW[i] = We[i];
    __syncthreads();

    const int j    = t & 63;
    const int rloc = t >> 6;             // 4 rows per block-iteration
    const float bej = be[j];
    float acc = 0.f;
    for (int row = blockIdx.x * 4 + rloc; row < N_EDGES; row += gridDim.x * 4) {
        const float* xr = EX + row * EDGE_F;
        float h = 0.f;
#pragma unroll
        for (int k = 0; k < EDGE_F; ++k) h = fmaf(xr[k], sW[k * HID + j], h);
        if (row < N_NODES) {
            he_pre[row * HID + j] = h;   // needed for gather + self term later
        } else {
            acc += fmaxf(h + bej, 0.f);  // tail rows: self loop only, fold into mean
        }
    }
    sred[t] = acc;
    __syncthreads();
    if (rloc == 0) {
        float s = sred[j] + sred[64 + j] + sred[128 + j] + sred[192 + j];
        atomicAdd(&pe_sum[j], s);
    }
}

// ---------------- combined scatter for both convs: one thread per (edge, feature) ----------------
__global__ void k_scatter(const int* __restrict__ srcv, const int* __restrict__ dstv,
                          const float* __restrict__ dinv,
                          const float* __restrict__ hn_pre, const float* __restrict__ he_pre,
                          float* __restrict__ hn_agg, float* __restrict__ he_agg) {
    long long idx = (long long)blockIdx.x * blockDim.x + threadIdx.x;
    if (idx >= (long long)N_EDGES * HID) return;
    const int e = (int)(idx >> 6);
    const int j = (int)(idx & 63);
    const int s = srcv[e];
    const int d = dstv[e];
    const float norm = dinv[s] * dinv[d];
    atomicAdd(&hn_agg[d * HID + j], norm * hn_pre[s * HID + j]);
    atomicAdd(&he_agg[d * HID + j], norm * he_pre[s * HID + j]);
}

// ---------------- finalize head rows: add self term + bias, relu, column sums ----------------
__global__ void k_finalize(const float* __restrict__ dinv,
                           const float* __restrict__ hn_pre, const float* __restrict__ he_pre,
                           const float* __restrict__ hn_agg, const float* __restrict__ he_agg,
                           const float* __restrict__ bn, const float* __restrict__ be,
                           float* __restrict__ pn_sum, float* __restrict__ pe_sum) {
    __shared__ float sn[256], se[256];
    const int t = threadIdx.x;
    const int j = t & 63;
    const int rloc = t >> 6;
    const float bnj = bn[j], bej = be[j];
    float an = 0.f, ae = 0.f;
    for (int i = blockIdx.x * 4 + rloc; i < N_NODES; i += gridDim.x * 4) {
        const float di = dinv[i];
        const float s2 = di * di;                       // self-loop norm = 1/deg
        float vn = hn_agg[i * HID + j] + s2 * hn_pre[i * HID + j] + bnj;
        float ve = he_agg[i * HID + j] + s2 * he_pre[i * HID + j] + bej;
        an += fmaxf(vn, 0.f);
        ae += fmaxf(ve, 0.f);
    }
    sn[t] = an; se[t] = ae;
    __syncthreads();
    if (rloc == 0) {
        atomicAdd(&pn_sum[j], sn[j] + sn[64 + j] + sn[128 + j] + sn[192 + j]);
        atomicAdd(&pe_sum[j], se[j] + se[64 + j] + se[128 + j] + se[192 + j]);
    }
}

// ---------------- head: means -> fc -> softmax ----------------
__global__ void k_head(const float* __restrict__ pn_sum, const float* __restrict__ pe_sum,
                       const float* __restrict__ Wfc, const float* __restrict__ bfc,
                       float* __restrict__ out) {
    __shared__ float x[2 * HID];
    __shared__ float logits[N_ACT];
    const int t = threadIdx.x;              // 128 threads
    x[t] = (t < HID) ? pn_sum[t] * (1.0f / N_NODES)
                     : pe_sum[t - HID] * (1.0f / N_EDGES);
    __syncthreads();
    if (t < N_ACT) {
        float s = bfc[t];
        for (int k = 0; k < 2 * HID; ++k) s = fmaf(x[k], Wfc[k * N_ACT + t], s);
        logits[t] = s;
    }
    __syncthreads();
    if (t == 0) {
        float mx = logits[0];
        for (int a = 1; a < N_ACT; ++a) mx = fmaxf(mx, logits[a]);
        float den = 0.f, ex[N_ACT];
        for (int a = 0; a < N_ACT; ++a) { ex[a] = __expf(logits[a] - mx); den += ex[a]; }
        for (int a = 0; a < N_ACT; ++a) out[a] = ex[a] / den;
    }
}

extern "C" void kernel_launch(void* const* d_in, const int* in_sizes, int n_in,
                              void* d_out, int out_size, void* d_ws, size_t ws_size,
                              hipStream_t stream) {
    const float* node_x = (const float*)d_in[0];
    const int*   eidx   = (const int*)  d_in[1];   // [2, N_EDGES] int32
    const float* edge_x = (const float*)d_in[2];
    const float* Wn     = (const float*)d_in[3];
    const float* bn     = (const float*)d_in[4];
    const float* We     = (const float*)d_in[5];
    const float* be     = (const float*)d_in[6];
    const float* Wfc    = (const float*)d_in[7];
    const float* bfc    = (const float*)d_in[8];
    const int* srcv = eidx;
    const int* dstv = eidx + N_EDGES;
    float* out = (float*)d_out;

    // workspace layout (floats): ~51.4 MB
    float* ws     = (float*)d_ws;
    float* deg    = ws;                         // 50048 (dinv after k_rsqrt)
    float* hn_pre = deg    + 50048;             // N_NODES*HID
    float* he_pre = hn_pre + N_NODES * HID;     // N_NODES*HID (head rows only)
    float* hn_agg = he_pre + N_NODES * HID;     // N_NODES*HID
    float* he_agg = hn_agg + N_NODES * HID;     // N_NODES*HID
    float* sums   = he_agg + N_NODES * HID;     // 128: [pn_sum | pe_sum]

    k_init<<<(N_NODES * HID + 255) / 256, 256, 0, stream>>>(deg, hn_agg, he_agg, sums);
    k_degree<<<(N_EDGES + 255) / 256, 256, 0, stream>>>(dstv, deg);
    k_rsqrt<<<(N_NODES + 255) / 256, 256, 0, stream>>>(deg);
    k_gemm_node<<<(M_TILES + 1) / 2, dim3(32, 8), 0, stream>>>(node_x, Wn, hn_pre);
    k_edge_mlp<<<2000, 256, 0, stream>>>(edge_x, We, be, he_pre, sums + HID);
    {
        long long total = (long long)N_EDGES * HID;
        int blocks = (int)((total + 255) / 256);
        k_scatter<<<blocks, 256, 0, stream>>>(srcv, dstv, deg, hn_pre, he_pre, hn_agg, he_agg);
    }
    k_finalize<<<1024, 256, 0, stream>>>(deg, hn_pre, he_pre, hn_agg, he_agg,
                                         bn, be, sums, sums + HID);
    k_head<<<1, 128, 0, stream>>>(sums, sums + HID, Wfc, bfc, out);
}